// MMGCN_64141041599009
// MI455X (gfx1250) — compile-verified
//
#include <hip/hip_runtime.h>
#include <math.h>

typedef __attribute__((ext_vector_type(2))) float v2f;
typedef __attribute__((ext_vector_type(8))) float v8f;

#define NN 50000
#define FF 64
#define DD 128
#define MM 3
#define EE 600000
#define CC 128  // both GCN layers output 128 channels

// ---------------- utility ----------------
__global__ void fill_zero_kernel(float* p, int n) {
    int i = blockIdx.x * blockDim.x + threadIdx.x;
    if (i < n) p[i] = 0.0f;
}

// deg[col[e]] += ew[e]; plus self-loop weight 1 at each node
__global__ void deg_kernel(const int* __restrict__ col, const float* __restrict__ ew,
                           float* __restrict__ deg, int nE, int nN) {
    int i = blockIdx.x * blockDim.x + threadIdx.x;
    if (i < nE) {
        atomicAdd(&deg[col[i]], ew[i]);
    } else if (i < nE + nN) {
        atomicAdd(&deg[i - nE], 1.0f);
    }
}

// deg -> dis = deg>0 ? rsqrt(deg) : 0   (in place)
__global__ void dis_kernel(float* deg, int n) {
    int i = blockIdx.x * blockDim.x + threadIdx.x;
    if (i < n) {
        float d = deg[i];
        deg[i] = (d > 0.0f) ? rsqrtf(d) : 0.0f;
    }
}

// out[n,c] = bias[c] + dis[n]^2 * H[n,c]   (self-loop term + bias as init)
__global__ void scatter_init_kernel(const float* __restrict__ H, const float* __restrict__ dis,
                                    const float* __restrict__ bias, float* __restrict__ out, int total) {
    int i = blockIdx.x * blockDim.x + threadIdx.x;
    if (i < total) {
        int node = i >> 7;
        int c = i & 127;
        float di = dis[node];
        out[i] = bias[c] + di * di * H[i];
    }
}

// per (edge, 4-feature chunk): out[col,:] += H[row,:] * dis[row]*ew*dis[col]
__global__ void scatter_edge_kernel(const int* __restrict__ row, const int* __restrict__ col,
                                    const float* __restrict__ ew, const float* __restrict__ dis,
                                    const float* __restrict__ H, float* __restrict__ out, int nE) {
    int i = blockIdx.x * blockDim.x + threadIdx.x;
    if (i >= nE * 32) return;
    int e  = i >> 5;
    int c4 = (i & 31) << 2;
    int r = row[e];
    int c = col[e];
    float norm = dis[r] * ew[e] * dis[c];
    const float4 h = *(const float4*)(H + (size_t)r * CC + c4);
    float* o = out + (size_t)c * CC + c4;
    atomicAdd(o + 0, h.x * norm);
    atomicAdd(o + 1, h.y * norm);
    atomicAdd(o + 2, h.z * norm);
    atomicAdd(o + 3, h.w * norm);
}

// per-channel sum & sumsq into gstats[0..127]=sum, [128..255]=sumsq
__global__ void bn_stats_kernel(const float* __restrict__ x, float* __restrict__ gstats, int total) {
    __shared__ float ls[256];
    if (threadIdx.x < 256) ls[threadIdx.x] = 0.0f;
    __syncthreads();
    for (int i = blockIdx.x * blockDim.x + threadIdx.x; i < total; i += gridDim.x * blockDim.x) {
        float v = x[i];
        int c = i & 127;
        atomicAdd(&ls[c], v);
        atomicAdd(&ls[128 + c], v * v);
    }
    __syncthreads();
    if (threadIdx.x < 256) atomicAdd(&gstats[threadIdx.x], ls[threadIdx.x]);
}

// x = relu((x - mu) * rsqrt(var + eps)), biased var, in place
__global__ void bn_apply_kernel(float* __restrict__ x, const float* __restrict__ gstats,
                                int total, float invN) {
    int i = blockIdx.x * blockDim.x + threadIdx.x;
    if (i < total) {
        int c = i & 127;
        float mu  = gstats[c] * invN;
        float var = gstats[128 + c] * invN - mu * mu;
        float v = (x[i] - mu) * rsqrtf(var + 1e-5f);
        x[i] = (v > 0.0f) ? v : 0.0f;
    }
}

// block-reduced grid-stride sum -> atomicAdd(dst)
__global__ void sum_kernel(const float* __restrict__ x, float* __restrict__ dst, int total) {
    __shared__ float sm[256];
    float acc = 0.0f;
    for (int i = blockIdx.x * blockDim.x + threadIdx.x; i < total; i += gridDim.x * blockDim.x)
        acc += x[i];
    sm[threadIdx.x] = acc;
    __syncthreads();
    for (int s = 128; s > 0; s >>= 1) {
        if (threadIdx.x < s) sm[threadIdx.x] += sm[threadIdx.x + s];
        __syncthreads();
    }
    if (threadIdx.x == 0) atomicAdd(dst, sm[0]);
}

// SE attention: y = sigmoid(relu(s @ fc1^T + b1) @ fc2^T + b2), s[m] = mean of branch m
__global__ void attention_kernel(const float* __restrict__ ssum,
                                 const float* __restrict__ fc1w, const float* __restrict__ fc1b,
                                 const float* __restrict__ fc2w, const float* __restrict__ fc2b,
                                 float* __restrict__ yv) {
    float s[MM];
    for (int m = 0; m < MM; ++m) s[m] = ssum[m] / (float)((size_t)NN * DD);
    float h[6 * MM];
    for (int j = 0; j < 6 * MM; ++j) {
        float t = fc1b[j];
        for (int m = 0; m < MM; ++m) t += s[m] * fc1w[j * MM + m];
        h[j] = (t > 0.0f) ? t : 0.0f;
    }
    for (int m = 0; m < MM; ++m) {
        float t = fc2b[m];
        for (int j = 0; j < 6 * MM; ++j) t += h[j] * fc2w[m * 6 * MM + j];
        yv[m] = 1.0f / (1.0f + expf(-t));
    }
}

// ---------------- WMMA f32 GEMM: H[N x C] = A[N x K] @ B[K x C] ----------------
// One 16x16 output tile per wave, K stepped by 4 via V_WMMA_F32_16X16X4_F32.
__global__ __launch_bounds__(256) void gemm_wmma_kernel(const float* __restrict__ A,
                                                        const float* __restrict__ B,
                                                        float* __restrict__ H,
                                                        int K, int tilesN, int C) {
    int wave = threadIdx.x >> 5;
    int tile = blockIdx.x * 8 + wave;
    int tm = tile / tilesN;
    int tn = tile % tilesN;
    int lane = threadIdx.x & 31;
    int half = lane >> 4;
    int l16  = lane & 15;
    int row = tm * 16 + l16;   // A row (M) held by this lane
    int col = tn * 16 + l16;   // B col (N) held by this lane

    v8f acc = {0.f, 0.f, 0.f, 0.f, 0.f, 0.f, 0.f, 0.f};
    const float* arow = A + (size_t)row * K + 2 * half;       // A[row, k0+2*half + {0,1}]
    const float* bcol = B + (size_t)(2 * half) * C + col;     // B[k0+2*half + {0,1}, col]
    for (int k0 = 0; k0 < K; k0 += 4) {
        v2f a = *(const v2f*)(arow + k0);
        v2f b;
        b[0] = bcol[(size_t)k0 * C];
        b[1] = bcol[(size_t)k0 * C + C];
        acc = __builtin_amdgcn_wmma_f32_16x16x4_f32(false, a, false, b, (short)0, acc, false, false);
    }
#pragma unroll
    for (int v = 0; v < 8; ++v)
        H[(size_t)(tm * 16 + v + 8 * half) * C + tn * 16 + l16] = acc[v];
}

// ---------------- final conv: out[n,o] = sum_{m,d} X2[m][n,d]*y[m]*W[o, m*128+d] + bias[o] ----
__global__ __launch_bounds__(256) void finalconv_wmma_kernel(const float* __restrict__ X2,
                                                             const float* __restrict__ W,   // [128][384]
                                                             const float* __restrict__ bias,
                                                             const float* __restrict__ yv,
                                                             float* __restrict__ out) {
    int wave = threadIdx.x >> 5;
    int tile = blockIdx.x * 8 + wave;
    int tm = tile / (DD / 16);
    int tn = tile % (DD / 16);
    int lane = threadIdx.x & 31;
    int half = lane >> 4;
    int l16  = lane & 15;
    int row = tm * 16 + l16;   // node n
    int col = tn * 16 + l16;   // output channel o
    float y0 = yv[0], y1 = yv[1], y2 = yv[2];

    v8f acc = {0.f, 0.f, 0.f, 0.f, 0.f, 0.f, 0.f, 0.f};
    for (int k0 = 0; k0 < MM * DD; k0 += 4) {
        int m = k0 >> 7;                     // constant over the 4-wide k step (128 % 4 == 0)
        int d = (k0 & 127) + 2 * half;
        float ym = (m == 0) ? y0 : ((m == 1) ? y1 : y2);
        v2f a = *(const v2f*)(X2 + ((size_t)m * NN + row) * DD + d);
        a *= ym;
        v2f b;
        b[0] = W[(size_t)col * (MM * DD) + k0 + 2 * half];
        b[1] = W[(size_t)col * (MM * DD) + k0 + 2 * half + 1];
        acc = __builtin_amdgcn_wmma_f32_16x16x4_f32(false, a, false, b, (short)0, acc, false, false);
    }
    float bo = bias[col];
#pragma unroll
    for (int v = 0; v < 8; ++v)
        out[(size_t)(tm * 16 + v + 8 * half) * DD + tn * 16 + l16] = acc[v] + bo;
}

// ---------------- host ----------------
extern "C" void kernel_launch(void* const* d_in, const int* in_sizes, int n_in,
                              void* d_out, int out_size, void* d_ws, size_t ws_size,
                              hipStream_t stream) {
    const float* feat = (const float*)d_in[0];
    const int*   adj  = (const int*)d_in[1];
    const float* aw   = (const float*)d_in[2];
    const float* w1   = (const float*)d_in[3];
    const float* b1   = (const float*)d_in[4];
    const float* w2   = (const float*)d_in[5];
    const float* b2   = (const float*)d_in[6];
    const float* fc1w = (const float*)d_in[7];
    const float* fc1b = (const float*)d_in[8];
    const float* fc2w = (const float*)d_in[9];
    const float* fc2b = (const float*)d_in[10];
    const float* cnnw = (const float*)d_in[11];
    const float* cnnb = (const float*)d_in[12];

    float* X2    = (float*)d_ws;                          // 3 * N * 128
    float* deg   = X2 + (size_t)MM * NN * DD;             // N (holds dis after dis_kernel)
    float* stats = deg + NN;                              // 256
    float* ssum  = stats + 256;                           // 3 (+pad)
    float* yv    = ssum + 8;                              // 3
    float* Hbuf  = (float*)d_out;                         // N*128 scratch; final output written last

    const dim3 blk(256);
    const int totalNC = NN * CC;
    const dim3 gNC((totalNC + 255) / 256);
    const dim3 gN((NN + 255) / 256);
    const dim3 gDeg((EE + NN + 255) / 256);
    const dim3 gEdge((EE * 32 + 255) / 256);
    const dim3 gGemm((NN / 16) * (CC / 16) / 8);          // 8 waves/block, 1 tile/wave
    const float invN = 1.0f / (float)NN;

    fill_zero_kernel<<<dim3(1), blk, 0, stream>>>(ssum, 16);

    for (int m = 0; m < MM; ++m) {
        const float* Xm = feat + (size_t)m * NN * FF;
        const int* row = adj + (size_t)m * 2 * EE;
        const int* col = row + EE;
        const float* ew = aw + (size_t)m * EE;
        float* X2m = X2 + (size_t)m * NN * DD;

        // H1 = Xm @ W1
        gemm_wmma_kernel<<<gGemm, blk, 0, stream>>>(Xm, w1, Hbuf, FF, CC / 16, CC);

        // degree -> dis (per branch edge set)
        fill_zero_kernel<<<gN, blk, 0, stream>>>(deg, NN);
        deg_kernel<<<gDeg, blk, 0, stream>>>(col, ew, deg, EE, NN);
        dis_kernel<<<gN, blk, 0, stream>>>(deg, NN);

        // GCN1 aggregate: X2m = b1 + dis^2*H1 + scatter(edges)
        scatter_init_kernel<<<gNC, blk, 0, stream>>>(Hbuf, deg, b1, X2m, totalNC);
        scatter_edge_kernel<<<gEdge, blk, 0, stream>>>(row, col, ew, deg, Hbuf, X2m, EE);

        // BN + ReLU (in place)
        fill_zero_kernel<<<dim3(1), blk, 0, stream>>>(stats, 256);
        bn_stats_kernel<<<dim3(512), blk, 0, stream>>>(X2m, stats, totalNC);
        bn_apply_kernel<<<gNC, blk, 0, stream>>>(X2m, stats, totalNC, invN);

        // H2 = X1 @ W2
        gemm_wmma_kernel<<<gGemm, blk, 0, stream>>>(X2m, w2, Hbuf, CC, CC / 16, CC);

        // GCN2 aggregate into X2m (X1 no longer needed)
        scatter_init_kernel<<<gNC, blk, 0, stream>>>(Hbuf, deg, b2, X2m, totalNC);
        scatter_edge_kernel<<<gEdge, blk, 0, stream>>>(row, col, ew, deg, Hbuf, X2m, EE);

        // BN + ReLU (in place)
        fill_zero_kernel<<<dim3(1), blk, 0, stream>>>(stats, 256);
        bn_stats_kernel<<<dim3(512), blk, 0, stream>>>(X2m, stats, totalNC);
        bn_apply_kernel<<<gNC, blk, 0, stream>>>(X2m, stats, totalNC, invN);

        // branch mean numerator for SE attention
        sum_kernel<<<dim3(512), blk, 0, stream>>>(X2m, ssum + m, totalNC);
    }

    attention_kernel<<<dim3(1), dim3(1), 0, stream>>>(ssum, fc1w, fc1b, fc2w, fc2b, yv);

    // out[n,o] = sum_{m,d} X2[m][n,d]*y[m]*cnn_w[o,m,d] + cnn_b[o]
    finalconv_wmma_kernel<<<gGemm, blk, 0, stream>>>(X2, cnnw, cnnb, yv, (float*)d_out);
}